// SparseQueryAndGroup_63024350102237
// MI455X (gfx1250) — compile-verified
//
#include <hip/hip_runtime.h>

// Problem constants (from reference): B=8, N=4096, K=2048, M=8192, C=64
#define NPTS      4096
#define NSAMPLE   32
#define RAD2      0.01f
#define KQ        2048
#define CCH       64
#define NBATCH    8
#define THREADS   256
#define PTS_PER_THREAD (NPTS / THREADS)   // 16
#define OUTCH     (3 + CCH)               // 67
#define FEATT_BYTES ((size_t)NBATCH * NPTS * CCH * sizeof(float))  // 8 MB

// Convert a generic pointer to a __shared__ object into a 32-bit LDS byte
// address (what the async-to-LDS instruction's VDST operand wants).
__device__ __forceinline__ unsigned lds_off_of(const void* p) {
  return (unsigned)(unsigned long long)
      (__attribute__((address_space(3))) const void*)p;
}

// ---------------------------------------------------------------------------
// Pre-pass: transpose features (B,C,N) -> (B,N,C) so that the per-slot
// channel gather in the main kernel becomes two contiguous b128 loads per
// lane instead of 8 scattered dword gathers. 8 MB once vs ~1 GB of scattered
// sector traffic per call.
// ---------------------------------------------------------------------------
#define TRTILE 32
__global__ __launch_bounds__(256)
void feat_transpose_kernel(const float* __restrict__ feat,   // (B,C,N)
                           float* __restrict__ featT)        // (B,N,C)
{
  __shared__ float tile[TRTILE][TRTILE + 1];
  const int tx = threadIdx.x;              // 0..31
  const int ty = threadIdx.y;              // 0..7
  const int n0 = blockIdx.x * TRTILE;      // N/32 tiles
  const int c0 = blockIdx.y * TRTILE;      // C/32 tiles
  const int b  = blockIdx.z;

#pragma unroll
  for (int i = 0; i < 4; ++i) {
    const int c = c0 + ty + 8 * i;
    tile[ty + 8 * i][tx] = feat[((size_t)b * CCH + c) * NPTS + n0 + tx];
  }
  __syncthreads();
#pragma unroll
  for (int i = 0; i < 4; ++i) {
    const int n = n0 + ty + 8 * i;
    featT[((size_t)b * NPTS + n) * CCH + c0 + tx] = tile[tx][ty + 8 * i];
  }
}

// ---------------------------------------------------------------------------
// Main kernel: one 256-thread block per query row m.
// ---------------------------------------------------------------------------
__global__ __launch_bounds__(THREADS)
void sparse_query_group_kernel(const float* __restrict__ xyz,      // (B,N,3)
                               const float* __restrict__ new_xyz,  // (B,K,3)
                               const float* __restrict__ feat,     // (B,C,N)
                               const float* __restrict__ featT,    // (B,N,C) or null
                               const int*   __restrict__ indices,  // (M,2)
                               float*       __restrict__ out)      // (M,67,32)
{
  __shared__ float s_pts[NPTS * 3];   // 48 KB: xyz[b] staged via async DMA
  __shared__ int   s_slot[NSAMPLE];
  __shared__ int   s_wsum[8];
  __shared__ int   s_woff[8];
  __shared__ int   s_cnt;

  const int m    = blockIdx.x;
  const int t    = threadIdx.x;
  const int lane = t & 31;
  const int wid  = t >> 5;

  const int b  = indices[2 * m + 0];
  const int kq = indices[2 * m + 1];

  const float qx = new_xyz[(b * KQ + kq) * 3 + 0];
  const float qy = new_xyz[(b * KQ + kq) * 3 + 1];
  const float qz = new_xyz[(b * KQ + kq) * 3 + 2];

  // ---------------- async stage xyz[b] (48 KB) into LDS ----------------
  {
    const unsigned long long gbase =
        (unsigned long long)(xyz + (size_t)b * NPTS * 3);
    const unsigned lbase = lds_off_of(s_pts);
#pragma unroll
    for (int j = 0; j < 12; ++j) {
      const int chunk = j * THREADS + t;                 // 3072 x 16B chunks
      unsigned long long ga = gbase + (unsigned long long)chunk * 16ull;
      unsigned la = lbase + (unsigned)chunk * 16u;
      asm volatile("global_load_async_to_lds_b128 %0, %1, off"
                   :: "v"(la), "v"(ga) : "memory");
    }
    asm volatile("s_wait_asynccnt 0" ::: "memory");
  }
  __syncthreads();

  // ------- phase 1: in-radius flags for 16 consecutive points/thread -------
  unsigned flags = 0u;
#pragma unroll
  for (int i = 0; i < PTS_PER_THREAD; ++i) {
    const int p = t * PTS_PER_THREAD + i;
    const float dx = s_pts[p * 3 + 0] - qx;
    const float dy = s_pts[p * 3 + 1] - qy;
    const float dz = s_pts[p * 3 + 2] - qz;
    const float d2 = dx * dx + dy * dy + dz * dz;
    flags |= (d2 < RAD2) ? (1u << i) : 0u;
  }
  const int mycnt = __popc(flags);

  // wave32 inclusive scan of per-thread hit counts (order-preserving ranks)
  int incl = mycnt;
#pragma unroll
  for (int d = 1; d < 32; d <<= 1) {
    int o = __shfl_up(incl, d, 32);
    if (lane >= d) incl += o;
  }
  if (lane == 31) s_wsum[wid] = incl;
  __syncthreads();
  if (t == 0) {
    int acc = 0;
#pragma unroll
    for (int w = 0; w < 8; ++w) { s_woff[w] = acc; acc += s_wsum[w]; }
    s_cnt = acc;
  }
  __syncthreads();
  const int excl = incl - mycnt + s_woff[wid];

  // scatter the first-NSAMPLE (in scan order) point indices into slots
  if (flags != 0u && excl < NSAMPLE) {
    int r = excl;
#pragma unroll
    for (int i = 0; i < PTS_PER_THREAD; ++i) {
      if ((flags >> i) & 1u) {
        if (r < NSAMPLE) s_slot[r] = t * PTS_PER_THREAD + i;
        ++r;
      }
    }
  }
  __syncthreads();

  // padding rule: unused slots <- first hit index; no hits -> all zeros
  if (t < NSAMPLE) {
    const int cnt = (s_cnt < NSAMPLE) ? s_cnt : NSAMPLE;
    int v;
    if (cnt == 0) {
      v = 0;
    } else {
      const int first = s_slot[0];
      v = (t < cnt) ? s_slot[t] : first;
    }
    s_slot[t] = v;   // single wave: DS read above ordered before this write
  }
  __syncthreads();

  // ---------------- phase 2: grouping (coalesced 128B row stores) -----------
  const int s   = lane;          // sample slot
  const int g   = wid;           // channel group (8 channels per wave)
  const int pid = s_slot[s];

  float* orow = out + (size_t)m * OUTCH * NSAMPLE;

  if (g == 0) {
    // grouped_xyz - center, read from the LDS-staged tile (reuse #2)
    orow[0 * NSAMPLE + s] = s_pts[pid * 3 + 0] - qx;
    orow[1 * NSAMPLE + s] = s_pts[pid * 3 + 1] - qy;
    orow[2 * NSAMPLE + s] = s_pts[pid * 3 + 2] - qz;
  }

  if (featT != nullptr) {
    // transposed layout: 8 channels = 32 contiguous bytes per lane
    const float4* fr = (const float4*)(featT +
        ((size_t)b * NPTS + pid) * CCH + g * 8);
    const float4 f0 = fr[0];
    const float4 f1 = fr[1];
    const int c0 = 3 + g * 8;
    orow[(size_t)(c0 + 0) * NSAMPLE + s] = f0.x;
    orow[(size_t)(c0 + 1) * NSAMPLE + s] = f0.y;
    orow[(size_t)(c0 + 2) * NSAMPLE + s] = f0.z;
    orow[(size_t)(c0 + 3) * NSAMPLE + s] = f0.w;
    orow[(size_t)(c0 + 4) * NSAMPLE + s] = f1.x;
    orow[(size_t)(c0 + 5) * NSAMPLE + s] = f1.y;
    orow[(size_t)(c0 + 6) * NSAMPLE + s] = f1.z;
    orow[(size_t)(c0 + 7) * NSAMPLE + s] = f1.w;
  } else {
    // fallback: direct scattered gather from (B,C,N)
    const float* frow = feat + (size_t)b * CCH * NPTS + pid;
#pragma unroll
    for (int j = 0; j < 8; ++j) {
      const int c = g * 8 + j;
      orow[(size_t)(3 + c) * NSAMPLE + s] = frow[(size_t)c * NPTS];
    }
  }
}

extern "C" void kernel_launch(void* const* d_in, const int* in_sizes, int n_in,
                              void* d_out, int out_size, void* d_ws, size_t ws_size,
                              hipStream_t stream) {
  const float* xyz     = (const float*)d_in[0];
  const float* new_xyz = (const float*)d_in[1];
  const float* feat    = (const float*)d_in[2];
  const int*   indices = (const int*)d_in[3];
  float*       out     = (float*)d_out;

  const int M = in_sizes[3] / 2;   // indices is (M,2)

  const float* featT = nullptr;
  if (ws_size >= FEATT_BYTES) {
    dim3 tgrid(NPTS / TRTILE, CCH / TRTILE, NBATCH);   // (128, 2, 8)
    dim3 tblk(32, 8);
    feat_transpose_kernel<<<tgrid, tblk, 0, stream>>>(feat, (float*)d_ws);
    featT = (const float*)d_ws;
  }

  sparse_query_group_kernel<<<M, THREADS, 0, stream>>>(xyz, new_xyz, feat,
                                                       featT, indices, out);
}